// SlideGraphArch_3281355014583
// MI455X (gfx1250) — compile-verified
//
#include <hip/hip_runtime.h>

typedef __attribute__((ext_vector_type(2))) float v2f;
typedef __attribute__((ext_vector_type(8))) float v8f;

static constexpr int   HDIM   = 64;
static constexpr float BN_EPS = 1e-5f;

__device__ __forceinline__ unsigned fkey(float x) {
  unsigned b = __float_as_uint(x);
  return (b & 0x80000000u) ? ~b : (b | 0x80000000u);
}
__device__ __forceinline__ float fdec(unsigned k) {
  return (k & 0x80000000u) ? __uint_as_float(k & 0x7fffffffu)
                           : __uint_as_float(~k);
}

// ---------------------------------------------------------------------------
// GEMM: Hout[N,64] = (X [+ AGG]) @ W[K,64] + bias, fused per-column sum/sumsq
// for BatchNorm stats. One wave computes a 16x64 output tile using
// V_WMMA_F32_16X16X4_F32: A fragment (16x4 f32) is loaded once per K-step and
// reused across the four 16-wide N tiles (4 wmma per K-step).
//
// f32 WMMA fragment layouts (wave32):
//   A 16x4 : lanes 0-15 -> M=lane, {K=k0,k0+1}; lanes 16-31 -> M=lane-16, {K=k0+2,k0+3}
//   B 4x16 : lanes 0-15 -> N=lane, {K=k0,k0+1}; lanes 16-31 -> N=lane-16, {K=k0+2,k0+3}
//   C/D    : VGPR v -> M = v + (lane>=16 ? 8 : 0), N = lane & 15
// ---------------------------------------------------------------------------
template<int K, bool ADD>
__global__ __launch_bounds__(256)
void gemm_bias_stats(const float* __restrict__ X, const float* __restrict__ AGG,
                     const float* __restrict__ W, const float* __restrict__ bias,
                     float* __restrict__ Hout,
                     float* __restrict__ colsum, float* __restrict__ colsq,
                     int nRowBlocks)
{
  __shared__ float ls[HDIM];
  __shared__ float lq[HDIM];
  if (threadIdx.x < HDIM) { ls[threadIdx.x] = 0.f; lq[threadIdx.x] = 0.f; }
  __syncthreads();

  const int lane = threadIdx.x & 31;
  const int wave = threadIdx.x >> 5;
  const int rb   = blockIdx.x * 8 + wave;

  if (rb < nRowBlocks) {
    const int mrow = rb * 16 + (lane & 15);   // A row supplied by this lane
    const int koff = (lane >> 4) << 1;        // 0 or 2
    const int ncol = lane & 15;

    v8f acc[4] = {};
    const float* __restrict__ xrow = X + (size_t)mrow * K;
    const float* __restrict__ grow = ADD ? (AGG + (size_t)mrow * K) : nullptr;

    for (int k = 0; k < K; k += 4) {
      v2f a = *(const v2f*)(xrow + k + koff);
      if (ADD) {
        v2f g = *(const v2f*)(grow + k + koff);
        a.x += g.x; a.y += g.y;
      }
      const float* wr0 = W + (size_t)(k + koff) * HDIM;
      const float* wr1 = wr0 + HDIM;
#pragma unroll
      for (int t = 0; t < 4; ++t) {
        v2f b;
        b.x = wr0[t * 16 + ncol];
        b.y = wr1[t * 16 + ncol];
        acc[t] = __builtin_amdgcn_wmma_f32_16x16x4_f32(
            /*neg_a=*/false, a, /*neg_b=*/false, b,
            /*c_mod=*/(short)0, acc[t], /*reuse_a=*/false, /*reuse_b=*/false);
      }
    }

    const int rhi = (lane >> 4) * 8;
#pragma unroll
    for (int t = 0; t < 4; ++t) {
      const int   col = t * 16 + ncol;
      const float bs  = bias[col];
      float s = 0.f, q = 0.f;
#pragma unroll
      for (int v = 0; v < 8; ++v) {
        const int row = rb * 16 + rhi + v;
        float val = acc[t][v] + bs;
        Hout[(size_t)row * HDIM + col] = val;
        s += val;
        q += val * val;
      }
      atomicAdd(&ls[col], s);   // ds_add_f32
      atomicAdd(&lq[col], q);
    }
  }
  __syncthreads();
  if (threadIdx.x < HDIM) {
    atomicAdd(&colsum[threadIdx.x], ls[threadIdx.x]);
    atomicAdd(&colsq[threadIdx.x],  lq[threadIdx.x]);
  }
}

// ---------------------------------------------------------------------------
// BatchNorm (training stats) + ReLU, float4-vectorized over the [N,64] tensor.
// ---------------------------------------------------------------------------
__global__ void bn_relu_kernel(const float* __restrict__ Hraw,
                               const float* __restrict__ colsum,
                               const float* __restrict__ colsq,
                               const float* __restrict__ gamma,
                               const float* __restrict__ beta,
                               float* __restrict__ Fout, int N, int total4)
{
  int idx = blockIdx.x * blockDim.x + threadIdx.x;
  if (idx >= total4) return;
  const int   c0   = (idx * 4) & (HDIM - 1);
  const float invN = 1.0f / (float)N;
  float4 h = ((const float4*)Hraw)[idx];
  float hv[4] = {h.x, h.y, h.z, h.w};
  float o[4];
#pragma unroll
  for (int j = 0; j < 4; ++j) {
    int   c   = c0 + j;
    float mu  = colsum[c] * invN;
    float var = colsq[c] * invN - mu * mu;
    float sc  = gamma[c] * __frsqrt_rn(var + BN_EPS);
    float sh  = beta[c] - mu * sc;
    o[j] = fmaxf(0.f, hv[j] * sc + sh);
  }
  ((float4*)Fout)[idx] = make_float4(o[0], o[1], o[2], o[3]);
}

// ---------------------------------------------------------------------------
// agg[dst] += feature[src] over E edges. float4 gather (feature table is
// 25.6 MB -> resident in the 192 MB L2), 4x global_atomic_add_f32 scatter.
// ---------------------------------------------------------------------------
__global__ void scatter_add_kernel(const float* __restrict__ Fin,
                                   const int* __restrict__ src,
                                   const int* __restrict__ dst,
                                   float* __restrict__ agg, int total16)
{
  int idx = blockIdx.x * blockDim.x + threadIdx.x;
  if (idx >= total16) return;
  int e = idx >> 4;
  int c = (idx & 15) << 2;
  int s = src[e], d = dst[e];
  float4 f = *(const float4*)(Fin + (size_t)s * HDIM + c);
  float* p = agg + (size_t)d * HDIM + c;
  atomicAdd(p + 0, f.x);
  atomicAdd(p + 1, f.y);
  atomicAdd(p + 2, f.z);
  atomicAdd(p + 3, f.w);
}

// ---------------------------------------------------------------------------
// Per-layer readout: sub = feature @ linW[64,2] + linb; node_pred (+)= sub;
// segment-max via monotone uint keys + atomicMax (handles negative floats).
// ---------------------------------------------------------------------------
template<bool FIRST>
__global__ void readout_kernel(const float* __restrict__ Fin,
                               const float* __restrict__ linW,
                               const float* __restrict__ linb,
                               const int* __restrict__ batch,
                               float* __restrict__ node_pred,
                               unsigned* __restrict__ wsikey, int N)
{
  int i = blockIdx.x * blockDim.x + threadIdx.x;
  if (i >= N) return;
  float s0 = linb[0], s1 = linb[1];
  const float4* fr = (const float4*)(Fin + (size_t)i * HDIM);
#pragma unroll
  for (int j = 0; j < 16; ++j) {
    float4 f = fr[j];
    const float* w = linW + j * 8;
    s0 += f.x * w[0] + f.y * w[2] + f.z * w[4] + f.w * w[6];
    s1 += f.x * w[1] + f.y * w[3] + f.z * w[5] + f.w * w[7];
  }
  if (FIRST) {
    node_pred[2 * i]     = s0;
    node_pred[2 * i + 1] = s1;
  } else {
    node_pred[2 * i]     += s0;
    node_pred[2 * i + 1] += s1;
  }
  int g = batch[i];
  atomicMax(&wsikey[2 * g + 0], fkey(s0));
  atomicMax(&wsikey[2 * g + 1], fkey(s1));
}

__global__ void init_keys_kernel(unsigned* wsikey) {
  if (threadIdx.x < 16) wsikey[threadIdx.x] = 0x007FFFFFu;  // key(-inf)
}

// Fold this layer's segment-max into the running wsi sum, emit current total,
// re-arm the keys for the next layer.
__global__ void wsi_accum_kernel(unsigned* wsikey, float* wsiacc, float* out) {
  int i = threadIdx.x;
  if (i < 16) {
    float v = wsiacc[i] + fdec(wsikey[i]);
    wsiacc[i] = v;
    out[i]    = v;
    wsikey[i] = 0x007FFFFFu;
  }
}

// ---------------------------------------------------------------------------
extern "C" void kernel_launch(void* const* d_in, const int* in_sizes, int n_in,
                              void* d_out, int out_size, void* d_ws, size_t ws_size,
                              hipStream_t stream)
{
  const float* x      = (const float*)d_in[0];
  const int*   eidx   = (const int*)  d_in[1];   // [2,E]: src then dst
  const int*   batch  = (const int*)  d_in[2];
  const float* W0     = (const float*)d_in[3];   // [128,64]
  const float* b0     = (const float*)d_in[4];
  const float* g0     = (const float*)d_in[5];
  const float* be0    = (const float*)d_in[6];
  const float* convW  = (const float*)d_in[7];   // [3,64,64]
  const float* convB  = (const float*)d_in[8];
  const float* convG  = (const float*)d_in[9];
  const float* convBe = (const float*)d_in[10];
  const float* linW   = (const float*)d_in[11];  // [4,64,2]
  const float* linb   = (const float*)d_in[12];  // [4,2]

  const int N = in_sizes[0] / 128;
  const int E = in_sizes[1] / 2;
  const int L = 3;

  float* out       = (float*)d_out;
  float* wsi_out   = out;        // [8,2] = 16 floats
  float* node_pred = out + 16;   // [N,2]

  float*    h_raw   = (float*)d_ws;
  float*    feature = h_raw   + (size_t)N * HDIM;
  float*    agg     = feature + (size_t)N * HDIM;
  float*    colsum  = agg     + (size_t)N * HDIM;
  float*    colsq   = colsum + HDIM;
  float*    wsiacc  = colsq  + HDIM;
  unsigned* wsikey  = (unsigned*)(wsiacc + 16);

  const int nRB        = N / 16;             // 6250 row tiles
  const int gemmBlocks = (nRB + 7) / 8;      // 8 waves / block
  const int total4     = N * HDIM / 4;
  const int total16    = E * 16;

  // Zero all accumulators every call (deterministic across graph replays).
  hipMemsetAsync(colsum, 0, (HDIM + HDIM + 16) * sizeof(float), stream);
  init_keys_kernel<<<1, 32, 0, stream>>>(wsikey);

  // first_h: Linear(128,64) + BN + ReLU
  gemm_bias_stats<128, false><<<gemmBlocks, 256, 0, stream>>>(
      x, nullptr, W0, b0, h_raw, colsum, colsq, nRB);
  bn_relu_kernel<<<(total4 + 255) / 256, 256, 0, stream>>>(
      h_raw, colsum, colsq, g0, be0, feature, N, total4);

  // layer-0 readout + global_max_pool
  readout_kernel<true><<<(N + 255) / 256, 256, 0, stream>>>(
      feature, linW, linb, batch, node_pred, wsikey, N);
  wsi_accum_kernel<<<1, 32, 0, stream>>>(wsikey, wsiacc, wsi_out);

  for (int l = 0; l < L; ++l) {
    hipMemsetAsync(agg, 0, (size_t)N * HDIM * sizeof(float), stream);
    hipMemsetAsync(colsum, 0, 2 * HDIM * sizeof(float), stream);

    scatter_add_kernel<<<(total16 + 255) / 256, 256, 0, stream>>>(
        feature, eidx, eidx + E, agg, total16);

    gemm_bias_stats<64, true><<<gemmBlocks, 256, 0, stream>>>(
        feature, agg, convW + (size_t)l * HDIM * HDIM, convB + l * HDIM,
        h_raw, colsum, colsq, nRB);
    bn_relu_kernel<<<(total4 + 255) / 256, 256, 0, stream>>>(
        h_raw, colsum, colsq, convG + l * HDIM, convBe + l * HDIM, feature, N, total4);

    readout_kernel<false><<<(N + 255) / 256, 256, 0, stream>>>(
        feature, linW + (size_t)(l + 1) * HDIM * 2, linb + (l + 1) * 2,
        batch, node_pred, wsikey, N);
    wsi_accum_kernel<<<1, 32, 0, stream>>>(wsikey, wsiacc, wsi_out);
  }
}